// ValueIterationNetwork_70970039599284
// MI455X (gfx1250) — compile-verified
//
#include <hip/hip_runtime.h>

// Value Iteration Network step for MI455X (gfx1250, wave32).
// Per step:  rv = v + rewards;  dyn = conv3x3(rv, W[90]);  softmax over taps
// per action;  q = adaptive 9-tap conv of rv;  v' = max_a q.
// Conv mapped to V_WMMA_F32_16X16X4_F32: M = 16 pixels, N = 96 channels
// (90 padded), K = 12 (9 taps padded) -> 3 WMMA k-steps x 6 n-tiles per wave.
// All padding is handled arithmetically (clamped loads + v_cndmask) so the
// WMMA inner loop has no divergent exec regions.

constexpr int kDepth   = 16;
constexpr int kB       = 16;
constexpr int kH       = 96;
constexpr int kW       = 96;
constexpr int kNPad    = 96;        // 10 actions * 9 taps = 90 -> pad to 96
constexpr int kKPad    = 12;        // 9 taps -> pad to 12 (3 WMMA k-steps)
constexpr int kWaves   = 6;         // 6 waves x 16 pixels = one 96-px row
constexpr int kThreads = kWaves * 32;

typedef __attribute__((ext_vector_type(2))) float v2f;
typedef __attribute__((ext_vector_type(8))) float v8f;

__global__ __launch_bounds__(kThreads)
void vin_step_kernel(const float* __restrict__ v_in,
                     const float* __restrict__ rewards,
                     const float* __restrict__ wt,   // [90][9] row-major
                     float* __restrict__ v_out) {
    // rv strip for rows h-1..h+1 with 1-col zero padding on each side
    __shared__ float lds_rv[3][kW + 2];
    // transposed + padded weights: [channel][tap], taps 9..11 and ch 90..95 = 0
    __shared__ float lds_w[kNPad][kKPad];
    // per-pixel transition logits for this row: [pixel][channel]
    __shared__ float lds_dyn[kW][kNPad];
    __shared__ float lds_red[kW];

    const int tid  = threadIdx.x;
    const int lane = tid & 31;
    const int wv   = tid >> 5;
    const int b    = blockIdx.x / kH;
    const int h    = blockIdx.x % kH;

    const float* vimg = v_in    + (size_t)b * kH * kW;
    const float* rimg = rewards + (size_t)b * kH * kW;

    // ---- Stage rv (v + rewards) rows h-1..h+1 into LDS, zero-padded ----
    // Branch-free: load clamped address, mask with cndmask.
    for (int i = tid; i < 3 * (kW + 2); i += kThreads) {
        const int r  = i / (kW + 2);
        const int c  = i % (kW + 2);
        const int hh = h - 1 + r;
        const int ww = c - 1;
        const bool inb = (hh >= 0) & (hh < kH) & (ww >= 0) & (ww < kW);
        const int hc = min(max(hh, 0), kH - 1);
        const int wc = min(max(ww, 0), kW - 1);
        const float val = vimg[hc * kW + wc] + rimg[hc * kW + wc];
        lds_rv[r][c] = inb ? val : 0.0f;
    }

    // ---- Stage padded transposed weights: lds_w[ch][k] ----
    for (int i = tid; i < kNPad * kKPad; i += kThreads) {
        const int ch = i / kKPad;
        const int k  = i % kKPad;
        const bool valid = (ch < 90) & (k < 9);
        const float val = wt[min(ch, 89) * 9 + min(k, 8)];   // unconditional
        lds_w[ch][k] = valid ? val : 0.0f;
    }
    __syncthreads();

    // ---- Build A fragments (16 pixels x 4 taps per k-step) ----
    // A layout (16x4 f32): lanes 0-15 hold {K=k0, k0+1} for row M=lane,
    //                      lanes 16-31 hold {K=k0+2, k0+3} for row M=lane-16.
    const int  m  = (lane & 15) + wv * 16;   // pixel within the row
    const bool lo = lane < 16;
#define RV(r, c) lds_rv[r][m + (c)]
    v2f afrag[3];
    afrag[0].x = lo ? RV(0, 0) : RV(0, 2);   // tap0 | tap2
    afrag[0].y = lo ? RV(0, 1) : RV(1, 0);   // tap1 | tap3
    afrag[1].x = lo ? RV(1, 1) : RV(2, 0);   // tap4 | tap6
    afrag[1].y = lo ? RV(1, 2) : RV(2, 1);   // tap5 | tap7
    afrag[2].x = lo ? RV(2, 2) : 0.0f;       // tap8 | pad
    afrag[2].y = 0.0f;                       // pad  | pad
#undef RV

    // ---- 6 n-tiles of 16 channels: B from lds_w, accumulate with WMMA ----
    const int n    = (lane & 15);            // channel within n-tile
    const int koff = lo ? 0 : 2;             // lanes 16-31 supply taps +2
#pragma unroll 1
    for (int nt = 0; nt < 6; ++nt) {
        const int ch = nt * 16 + n;
        // Contiguous {t0,t1} pairs -> single b64 LDS loads, no conditions.
        v2f b0 = *(const v2f*)&lds_w[ch][0 + koff];
        v2f b1 = *(const v2f*)&lds_w[ch][4 + koff];
        v2f b2 = *(const v2f*)&lds_w[ch][8 + koff];
        v8f acc = {};
        acc = __builtin_amdgcn_wmma_f32_16x16x4_f32(false, afrag[0], false, b0,
                                                    (short)0, acc, false, false);
        acc = __builtin_amdgcn_wmma_f32_16x16x4_f32(false, afrag[1], false, b1,
                                                    (short)0, acc, false, false);
        acc = __builtin_amdgcn_wmma_f32_16x16x4_f32(false, afrag[2], false, b2,
                                                    (short)0, acc, false, false);
        // D layout: VGPR r holds row M = r + 8*(lane>=16), col N = lane&15
        const int mbase = wv * 16 + (lo ? 0 : 8);
#pragma unroll
        for (int r = 0; r < 8; ++r)
            lds_dyn[mbase + r][ch] = acc[r];
    }
    __syncthreads();

    // ---- Softmax over 9 taps per action, adaptive conv, max over actions ---
    // 192 threads: pixel = tid%96; half = tid/96 covers 5 of the 10 actions.
    const int p    = tid % kW;
    const int half = tid / kW;

    float patch[9];
#pragma unroll
    for (int t = 0; t < 9; ++t)
        patch[t] = lds_rv[t / 3][p + (t % 3)];

    float best = -3.402823466e38f;
    for (int a = half * 5; a < half * 5 + 5; ++a) {
        const float* dl = &lds_dyn[p][a * 9];
        float mx = dl[0];
#pragma unroll
        for (int t = 1; t < 9; ++t) mx = fmaxf(mx, dl[t]);
        float ssum = 0.0f, q = 0.0f;
#pragma unroll
        for (int t = 0; t < 9; ++t) {
            const float e = __expf(dl[t] - mx);
            ssum += e;
            q    += e * patch[t];
        }
        best = fmaxf(best, q / ssum);
    }

    if (half == 1) lds_red[p] = best;
    __syncthreads();
    if (half == 0) {
        best = fmaxf(best, lds_red[p]);
        v_out[((size_t)b * kH + h) * kW + p] = best;
    }
}

extern "C" void kernel_launch(void* const* d_in, const int* in_sizes, int n_in,
                              void* d_out, int out_size, void* d_ws, size_t ws_size,
                              hipStream_t stream) {
    (void)in_sizes; (void)n_in; (void)out_size; (void)d_ws; (void)ws_size;
    const float* values  = (const float*)d_in[0];
    const float* rewards = (const float*)d_in[1];
    const float* wt      = (const float*)d_in[2];
    float* out = (float*)d_out;

    const size_t slice = (size_t)kB * kH * kW;   // one depth's value map
    for (int t = 0; t < kDepth; ++t) {
        const float* vin = (t == 0) ? values : out + (size_t)(t - 1) * slice;
        float* vout = out + (size_t)t * slice;
        vin_step_kernel<<<kB * kH, kThreads, 0, stream>>>(vin, rewards, wt, vout);
    }
}